// fqconv_ansatz_layer_5703716569534
// MI455X (gfx1250) — compile-verified
//
#include <hip/hip_runtime.h>
#include <math.h>

typedef __attribute__((ext_vector_type(2))) float v2f;
typedef __attribute__((ext_vector_type(8))) float v8f;

#define NQ   12
#define DIM  4096
#define NT   256   // threads per block = 8 wave32
#define APT  16    // amplitudes per thread

// One workgroup per circuit. State lives entirely in LDS (32KB), zero HBM
// traffic for the statevector. Gate tables: wire w maps to bit (11-w).
__global__ __launch_bounds__(NT)
void fqconv_ansatz_kernel(const float* __restrict__ x,
                          const float* __restrict__ wgt,
                          float* __restrict__ out)
{
    __shared__ float sre[DIM];          // real part; reused as probs at the end
    __shared__ float sim[DIM];          // imag part
    __shared__ float ic[NQ], isn[NQ];   // cos/sin(angle/2) per wire
    __shared__ float wc[24], ws[24];    // cos/sin(weight/2) per gate
    __shared__ float red[8][16][7];     // per-wave WMMA partials
    __shared__ float sarr[NT];          // per-thread prob sums (wires 6,7)

    const int tid = threadIdx.x;
    const int ci  = blockIdx.x;          // circuit index = g*512 + b
    const int g   = ci >> 9;
    const int b   = ci & 511;
    const float* ang = x + b * 48 + g * 12;

    if (tid == 0) __builtin_prefetch(ang, 0, 0);   // global_prefetch_b8

    // --- trig tables -------------------------------------------------------
    if (tid < NQ) {
        float s, c; sincosf(0.5f * ang[tid], &s, &c);
        ic[tid] = c; isn[tid] = s;
    }
    if (tid < 24) {
        float s, c; sincosf(0.5f * wgt[tid], &s, &c);
        wc[tid] = c; ws[tid] = s;
    }
    __syncthreads();

    // --- closed-form initial product state: RX(theta_w)|0> per wire -------
    // amp(j) = prod_w (bit_{11-w}(j) ? s_w : c_w) * (-i)^popcount(j)
    float Mh = 1.0f;
    #pragma unroll
    for (int w = 0; w < 8; ++w) {                 // wires 0..7 <-> tid bits 7..0
        int bit = (tid >> (7 - w)) & 1;
        Mh *= bit ? isn[w] : ic[w];
    }
    const int pch = __popc(tid);
    #pragma unroll
    for (int r = 0; r < APT; ++r) {
        int j = tid * APT + r;
        float M = Mh;
        #pragma unroll
        for (int w = 8; w < 12; ++w) {            // wires 8..11 <-> r bits 3..0
            int bit = (r >> (11 - w)) & 1;
            M *= bit ? isn[w] : ic[w];
        }
        int k = (pch + __popc(r)) & 3;            // (-i)^k
        float re, im;
        if      (k == 0) { re =  M; im = 0.f; }
        else if (k == 1) { re = 0.f; im = -M; }
        else if (k == 2) { re = -M; im = 0.f; }
        else             { re = 0.f; im =  M; }
        sre[j] = re; sim[j] = im;
    }
    __syncthreads();

    // --- 24 controlled rotations ------------------------------------------
    // kind: 0 = CRZ (diagonal phase), 1 = CRX (pair mix)
    const int KIND[24] = {0,1,0,1, 1,0,1,0, 0,1,0,1, 1,0,1,0, 0,1,0,1, 1,0,1,0};
    const int CB[24]   = {8,9,10,11, 8,9,10,11, 0,1,2,3, 0,1,2,3, 4,5,6,7, 4,5,6,7};
    const int TB[24]   = {0,1,2,3, 0,1,2,3, 4,5,6,7, 4,5,6,7, 8,9,10,11, 8,9,10,11};

    #pragma unroll
    for (int gt = 0; gt < 24; ++gt) {
        const int cb = CB[gt], tb = TB[gt];
        const float c = wc[gt], s = ws[gt];
        if (KIND[gt] == 0) {
            // CRZ: amp *= e^{-i s} (tgt=0) / e^{+i s} (tgt=1), ctrl bit = 1
            #pragma unroll
            for (int u = 0; u < 8; ++u) {
                int q = tid + u * NT;                         // 2048 ctrl=1 amps
                int j = ((q >> cb) << (cb + 1)) | (1u << cb) | (q & ((1u << cb) - 1));
                float pi = ((j >> tb) & 1) ? s : -s;
                float ar = sre[j], ai = sim[j];
                sre[j] = ar * c - ai * pi;
                sim[j] = ar * pi + ai * c;
            }
        } else {
            // CRX: mix (j0, j0|1<<tb) pairs with ctrl bit = 1
            const int lo = (cb < tb) ? cb : tb;
            const int hi = (cb < tb) ? tb : cb;
            #pragma unroll
            for (int u = 0; u < 4; ++u) {
                int q  = tid + u * NT;                        // 1024 pairs
                int t1 = ((q  >> lo) << (lo + 1)) | (q  & ((1u << lo) - 1));
                int t2 = ((t1 >> hi) << (hi + 1)) | (t1 & ((1u << hi) - 1));
                int j0 = t2 | (1u << cb);
                int j1 = j0 | (1u << tb);
                float a0r = sre[j0], a0i = sim[j0];
                float a1r = sre[j1], a1i = sim[j1];
                sre[j0] = c * a0r + s * a1i;
                sim[j0] = c * a0i - s * a1r;
                sre[j1] = c * a1r + s * a0i;
                sim[j1] = c * a1i - s * a0r;
            }
        }
        __syncthreads();
    }

    // --- probabilities (reuse sre[]) + per-thread sums for wires 6/7 ------
    float S = 0.f;
    #pragma unroll
    for (int r = 0; r < APT; ++r) {
        int j = tid * APT + r;
        float p = sre[j] * sre[j] + sim[j] * sim[j];
        S += p;
        sre[j] = p;
    }
    sarr[tid] = S;
    __syncthreads();

    // --- WMMA signed-marginal reduction -----------------------------------
    // D += A(16x4) * B(4x16) over 8 chunks/wave (64 chunks total).
    // A is K-uniform: row w<6 = sign of chunk bit (5-w)  -> wires 0..5
    //                 row 8   = 1.0 (column marginal)    -> wires 8..11
    // B columns n = prob index bits 3..0; the 4 K-slots cover bits 5..4
    // (any K permutation valid since A is K-uniform).
    const int lane = tid & 31;
    const int wave = tid >> 5;
    const int n16  = lane & 15;
    const int hih  = lane >> 4;          // 0: K={0,1} half, 1: K={2,3} half
    v8f acc = {0.f, 0.f, 0.f, 0.f, 0.f, 0.f, 0.f, 0.f};
    #pragma unroll
    for (int mm = 0; mm < 8; ++mm) {
        int m = wave * 8 + mm;
        float av;
        if (n16 < 6)       av = 1.f - 2.f * (float)((m >> (5 - n16)) & 1);
        else if (n16 == 8) av = 1.f;
        else               av = 0.f;
        v2f A; A.x = av; A.y = av;       // same value for both K columns
        int base = m * 64 + (hih ? 32 : 0) + n16;
        v2f B; B.x = sre[base]; B.y = sre[base + 16];
        acc = __builtin_amdgcn_wmma_f32_16x16x4_f32(
                  false, A, false, B, (short)0, acc, false, false);
    }
    // D layout: VGPR v, lanes 0-15 -> row v, lanes 16-31 -> row v+8
    if (!hih) {
        #pragma unroll
        for (int w = 0; w < 6; ++w) red[wave][n16][w] = acc[w];
    } else {
        red[wave][n16][6] = acc[0];      // row 8 (all-ones) column marginal
    }
    __syncthreads();

    // --- final expectation values (deterministic sums) --------------------
    if (tid < 12) {
        float E = 0.f;
        if (tid < 6) {                               // wires 0..5 via sign rows
            for (int w8 = 0; w8 < 8; ++w8)
                for (int n = 0; n < 16; ++n) E += red[w8][n][tid];
        } else if (tid < 8) {                        // wires 6,7 <-> tid bits 1,0
            int sel = (tid == 6) ? 2 : 1;
            for (int t = 0; t < NT; ++t) E += (t & sel) ? -sarr[t] : sarr[t];
        } else {                                     // wires 8..11 <-> n bits 3..0
            int bp = 11 - tid;
            for (int n = 0; n < 16; ++n) {
                float cs = 0.f;
                for (int w8 = 0; w8 < 8; ++w8) cs += red[w8][n][6];
                E += ((n >> bp) & 1) ? -cs : cs;
            }
        }
        out[ci * 12 + tid] = E;
    }
}

extern "C" void kernel_launch(void* const* d_in, const int* in_sizes, int n_in,
                              void* d_out, int out_size, void* d_ws, size_t ws_size,
                              hipStream_t stream) {
    const float* x = (const float*)d_in[0];           // [512,12,2,2] f32
    const float* w = (const float*)d_in[1];           // [24] f32
    float* out = (float*)d_out;                       // [512,12,2,2] f32
    fqconv_ansatz_kernel<<<dim3(2048), dim3(NT), 0, stream>>>(x, w, out);
}